// VariLoss_40166534152779
// MI455X (gfx1250) — compile-verified
//
#include <hip/hip_runtime.h>
#include <hip/hip_bf16.h>
#include <math.h>

typedef __attribute__((ext_vector_type(2))) float v2f;
typedef __attribute__((ext_vector_type(8))) float v8f;

#define LEN_EPS 1e-12f
#define SRNF_W  1e-7f
// SIG = 0.5 -> 1/SIG^2 = 4
#define INV_SIG2 4.0f

// ---------------------------------------------------------------------------
// Kernel 0: zero the two accumulators (vari sum, srvf sum)
// ---------------------------------------------------------------------------
__global__ void zero_acc_kernel(float* acc) {
    acc[0] = 0.0f;
    acc[1] = 0.0f;
}

// ---------------------------------------------------------------------------
// Kernel 1: per-(sample, edge) precompute of edge descriptors.
//   CA[s][e] = (cx,cy,cz,|c|^2) for X = output[s]+template, UA = (ux,uy,uz,l)
//   CB/UB    = same for target[s] via tedges
//   Also accumulates the SRVF term  sum((srvf(X)-srvf(template))^2) into acc[1].
//   Edges e in [E, Epad) are zero-filled (l=0 kills their contribution).
// ---------------------------------------------------------------------------
__global__ void edge_kernel(const float* __restrict__ outp,
                            const float* __restrict__ tgt,
                            const int*   __restrict__ tedges,
                            const float* __restrict__ tmpl,
                            const int*   __restrict__ edges,
                            float4* __restrict__ CA, float4* __restrict__ UA,
                            float4* __restrict__ CB, float4* __restrict__ UB,
                            float* __restrict__ acc,
                            int B, int N, int E, int Epad)
{
    int tid = blockIdx.x * blockDim.x + threadIdx.x;
    if (tid >= B * Epad) return;
    int s = tid / Epad;
    int e = tid - s * Epad;
    size_t o = (size_t)s * Epad + e;

    if (e >= E) {
        float4 z = make_float4(0.f, 0.f, 0.f, 0.f);
        CA[o] = z; UA[o] = z; CB[o] = z; UB[o] = z;
        return;
    }

    // ---- X = output[s] + template, edge (i0,i1) ----
    int i0 = edges[2 * e + 0];
    int i1 = edges[2 * e + 1];
    const float* os = outp + (size_t)s * N * 3;

    float ax = os[3 * i0 + 0] + tmpl[3 * i0 + 0];
    float ay = os[3 * i0 + 1] + tmpl[3 * i0 + 1];
    float az = os[3 * i0 + 2] + tmpl[3 * i0 + 2];
    float bx = os[3 * i1 + 0] + tmpl[3 * i1 + 0];
    float by = os[3 * i1 + 1] + tmpl[3 * i1 + 1];
    float bz = os[3 * i1 + 2] + tmpl[3 * i1 + 2];

    float cx = 0.5f * (ax + bx), cy = 0.5f * (ay + by), cz = 0.5f * (az + bz);
    float tx = bx - ax, ty = by - ay, tz = bz - az;
    float lA = sqrtf(tx * tx + ty * ty + tz * tz + LEN_EPS);
    float rA = 1.0f / lA;
    float uxA = tx * rA, uyA = ty * rA, uzA = tz * rA;

    CA[o] = make_float4(cx, cy, cz, cx * cx + cy * cy + cz * cz);
    UA[o] = make_float4(uxA, uyA, uzA, lA);

    // ---- target side, edge from tedges[s] ----
    int j0 = tedges[((size_t)s * E + e) * 2 + 0];
    int j1 = tedges[((size_t)s * E + e) * 2 + 1];
    const float* ts = tgt + (size_t)s * N * 3;

    float px = ts[3 * j0 + 0], py = ts[3 * j0 + 1], pz = ts[3 * j0 + 2];
    float qx = ts[3 * j1 + 0], qy = ts[3 * j1 + 1], qz = ts[3 * j1 + 2];
    float dx = qx - px, dy = qy - py, dz = qz - pz;
    float ccx = 0.5f * (px + qx), ccy = 0.5f * (py + qy), ccz = 0.5f * (pz + qz);
    float lB = sqrtf(dx * dx + dy * dy + dz * dz + LEN_EPS);
    float rB = 1.0f / lB;

    CB[o] = make_float4(ccx, ccy, ccz, ccx * ccx + ccy * ccy + ccz * ccz);
    UB[o] = make_float4(dx * rB, dy * rB, dz * rB, lB);

    // ---- SRVF term: (u_X*sqrt(l_X) - u_T*sqrt(l_T))^2 ----
    float t0x = tmpl[3 * i0 + 0], t0y = tmpl[3 * i0 + 1], t0z = tmpl[3 * i0 + 2];
    float t1x = tmpl[3 * i1 + 0], t1y = tmpl[3 * i1 + 1], t1z = tmpl[3 * i1 + 2];
    float wx = t1x - t0x, wy = t1y - t0y, wz = t1z - t0z;
    float lT = sqrtf(wx * wx + wy * wy + wz * wz + LEN_EPS);
    float rT = 1.0f / lT;
    float sA = sqrtf(lA), sT = sqrtf(lT);
    float q0 = uxA * sA - wx * rT * sT;
    float q1 = uyA * sA - wy * rT * sT;
    float q2 = uzA * sA - wz * rT * sT;
    atomicAdd(&acc[1], q0 * q0 + q1 * q1 + q2 * q2);
}

// ---------------------------------------------------------------------------
// Kernel 2: WMMA pairwise varifold sums.
//   One wave per (row-tile, term, sample). term: 0=aa(+1) 1=ab(-2) 2=bb(+1).
//   Per 16x16 tile: two V_WMMA_F32_16X16X4_F32 (center dots, unit dots),
//   then elementwise exp/weights, accumulate, wave-reduce, atomicAdd.
// ---------------------------------------------------------------------------
__global__ void vari_pair_kernel(const float4* __restrict__ CA,
                                 const float4* __restrict__ UA,
                                 const float4* __restrict__ CB,
                                 const float4* __restrict__ UB,
                                 float* __restrict__ acc,
                                 int Epad)
{
    const int lane   = threadIdx.x;      // 0..31 (wave32)
    const int tile_i = blockIdx.x;
    const int term   = blockIdx.y;
    const int s      = blockIdx.z;
    const int ntiles = Epad >> 4;

    const float4* CP; const float4* UP; const float4* CQ; const float4* UQ;
    float w;
    if (term == 0)      { CP = CA; UP = UA; CQ = CA; UQ = UA; w =  1.0f; }
    else if (term == 1) { CP = CA; UP = UA; CQ = CB; UQ = UB; w = -2.0f; }
    else                { CP = CB; UP = UB; CQ = CB; UQ = UB; w =  1.0f; }
    const size_t sb = (size_t)s * Epad;
    CP += sb; UP += sb; CQ += sb; UQ += sb;

    const int  row = lane & 15;
    const bool hi  = lane >= 16;         // K=2,3 half of the A/B striping
    const int  i0  = tile_i * 16;

    // A-operand: 16x4 f32, lanes 0-15 -> {K0,K1}, lanes 16-31 -> {K2, 0pad}
    float4 cp = CP[i0 + row];
    float4 up = UP[i0 + row];
    v2f a_c, a_u;
    a_c[0] = hi ? cp.z : cp.x;  a_c[1] = hi ? 0.0f : cp.y;
    a_u[0] = hi ? up.z : up.x;  a_u[1] = hi ? 0.0f : up.y;

    // Row norms/lengths for this lane's D-layout half: M = rowbase + v
    const int rowbase = hi ? 8 : 0;
    float pn[8], lp[8];
#pragma unroll
    for (int v = 0; v < 8; ++v) {
        pn[v] = CP[i0 + rowbase + v].w;
        lp[v] = UP[i0 + rowbase + v].w;
    }

    float accl = 0.0f;
    for (int tj = 0; tj < ntiles; ++tj) {
        const int j0 = tj * 16;
        // B-operand: 4x16 f32, column N = lane&15; symmetric K striping
        float4 cq = CQ[j0 + row];
        float4 uq = UQ[j0 + row];
        v2f b_c, b_u;
        b_c[0] = hi ? cq.z : cq.x;  b_c[1] = hi ? 0.0f : cq.y;
        b_u[0] = hi ? uq.z : uq.x;  b_u[1] = hi ? 0.0f : uq.y;
        const float qn = cq.w;   // |c_q|^2 for column N = lane&15
        const float lq = uq.w;   // l_q      for column N = lane&15

        v8f cz = {};
        v8f dc = __builtin_amdgcn_wmma_f32_16x16x4_f32(
            false, a_c, false, b_c, (short)0, cz, false, false);
        v8f du = __builtin_amdgcn_wmma_f32_16x16x4_f32(
            false, a_u, false, b_u, (short)0, cz, false, false);

#pragma unroll
        for (int v = 0; v < 8; ++v) {
            float d2  = pn[v] + qn - 2.0f * dc[v];
            float K   = __expf(-INV_SIG2 * d2);   // exp(-d2/sig^2), sig=0.5
            float ang = du[v] * du[v];
            accl += K * ang * (lp[v] * lq);
        }
    }

    // wave32 reduction
#pragma unroll
    for (int off = 16; off > 0; off >>= 1)
        accl += __shfl_xor(accl, off, 32);
    if (lane == 0) atomicAdd(&acc[0], w * accl);
}

// ---------------------------------------------------------------------------
// Kernel 3: finalize  loss = mean_s( vari_s + SRNF_W * srvf_s )
// ---------------------------------------------------------------------------
__global__ void finalize_kernel(const float* __restrict__ acc,
                                float* __restrict__ out, float invB) {
    out[0] = (acc[0] + SRNF_W * acc[1]) * invB;
}

// ---------------------------------------------------------------------------
extern "C" void kernel_launch(void* const* d_in, const int* in_sizes, int n_in,
                              void* d_out, int out_size, void* d_ws, size_t ws_size,
                              hipStream_t stream) {
    const float* outp   = (const float*)d_in[0];   // (B,N,3) f32
    const float* target = (const float*)d_in[1];   // (B,N,3) f32
    const int*   tedges = (const int*)d_in[2];     // (B,E,2) i32
    const float* tmpl   = (const float*)d_in[3];   // (N,3)   f32
    const int*   edges  = (const int*)d_in[4];     // (E,2)   i32

    const int ND   = in_sizes[3];
    const int N    = ND / 3;
    const int B    = in_sizes[0] / ND;
    const int E    = in_sizes[4] / 2;
    const int Epad = (E + 15) & ~15;

    float*  acc  = (float*)d_ws;
    float4* base = (float4*)((char*)d_ws + 256);
    float4* CA = base;
    float4* UA = CA + (size_t)B * Epad;
    float4* CB = UA + (size_t)B * Epad;
    float4* UB = CB + (size_t)B * Epad;

    zero_acc_kernel<<<1, 1, 0, stream>>>(acc);

    const int tot = B * Epad;
    edge_kernel<<<(tot + 255) / 256, 256, 0, stream>>>(
        outp, target, tedges, tmpl, edges, CA, UA, CB, UB, acc, B, N, E, Epad);

    dim3 grid(Epad / 16, 3, B);
    vari_pair_kernel<<<grid, 32, 0, stream>>>(CA, UA, CB, UB, acc, Epad);

    finalize_kernel<<<1, 1, 0, stream>>>(acc, (float*)d_out, 1.0f / (float)B);
}